// RiemannSolver_83820581749014
// MI455X (gfx1250) — compile-verified
//
#include <hip/hip_runtime.h>
#include <math.h>

// ---------------------------------------------------------------------------
// Riemann solver with MLP flux closure, CDNA5 (gfx1250) WMMA implementation.
// Each wave32 processes 32 cells; MLP matmuls run on V_WMMA_F32_16X16X4_F32.
// Weights and the hidden activations are stored in LDS pre-swizzled into the
// exact WMMA per-lane operand layout, so every A/B operand is one aligned
// ds_load_b64 at (base + lane*8 + imm): no repacking movs, no bank conflicts.
// ---------------------------------------------------------------------------

typedef __attribute__((ext_vector_type(2))) float v2f;
typedef __attribute__((ext_vector_type(8))) float v8f;

#define F_GAMMA  1.6666666666666667f
#define F_CUTOFF 0.005f

// --------------------------- LDS layout (floats) ---------------------------
// W1 swizzled:  [mlp][kb(5)][nb(4)][lane(32)][2]   -> 3*1280 = 3840
// W2 swizzled:  [mlp][kb(16)][lane(32)][2] (j>=3 zero-padded) -> 3*1024 = 3072
#define L_W1SW 0                    // 3840
#define L_B1   3840                 // 3*64 = 192
#define L_W2SW 4032                 // 3072
#define L_B2   7104                 // 9 -> pad 16
#define L_WAVE 7120
#define WV_FEAT 0                   // 32*20 = 640
#define WV_HTP  640                 // hidden pre-paired: [kb(16)][lane(32)][2] = 1024
#define WV_INFO 1664                // 32*6 = 192 (hlle0..2, label, cont, flip)
#define WV_FLUX 1856                // 32*3 = 96
#define WV_SIZE 1952
#define LDS_TOT (L_WAVE + 4 * WV_SIZE)   // 14928 floats = 59712 B

__device__ __forceinline__ v8f wmma_f32(v2f a, v2f b, v8f c) {
  // 8 args: (neg_a, A, neg_b, B, c_mod, C, reuse_a, reuse_b)
  return __builtin_amdgcn_wmma_f32_16x16x4_f32(false, a, false, b, (short)0, c,
                                               false, false);
}

__device__ __forceinline__ float fast_tanh(float x) {
#if defined(__HIP_DEVICE_COMPILE__) && __has_builtin(__builtin_amdgcn_tanhf)
  return __builtin_amdgcn_tanhf(x);   // V_TANH_F32 hardware transcendental
#elif defined(__HIP_DEVICE_COMPILE__) && __has_builtin(__builtin_amdgcn_tanh_f32)
  return __builtin_amdgcn_tanh_f32(x);
#else
  return tanhf(x);
#endif
}

// ------------------- per-cell scalar preprocessing -------------------------
struct CellPre {
  float feats[20];
  float hl[3];
  int   label;
  bool  cont;
  bool  flip;
};

__device__ __forceinline__ void cell_pre(const float* __restrict__ P,
                                         const float* __restrict__ U,
                                         const float* __restrict__ F,
                                         const float* __restrict__ cmax,
                                         const float* __restrict__ cmin,
                                         int cell, CellPre& o) {
  float Pl[6], Ul[6], Fl[6];
#pragma unroll
  for (int i = 0; i < 6; ++i) {
    Pl[i] = P[cell * 6 + i];
    Ul[i] = U[cell * 6 + i];
    Fl[i] = F[cell * 6 + i];
  }
  const bool flip = Pl[3] > Pl[2];  // P[1][1] > P[1][0]
  const float sg[3] = {1.f, 1.f, -1.f};
  float Pf[6], Uf[6], Ff[6];
#pragma unroll
  for (int k = 0; k < 3; ++k) {
#pragma unroll
    for (int s = 0; s < 2; ++s) {
      const int d = k * 2 + s, sc = k * 2 + (1 - s);
      Pf[d] = flip ? Pl[sc] * sg[k] : Pl[d];
      Uf[d] = flip ? Ul[sc] * sg[k] : Ul[d];
      Ff[d] = flip ? Fl[sc] * (-sg[k]) : Fl[d];
    }
  }
  const float dmax = fmaxf(fabsf(Pf[1] - Pf[0]),
                           fmaxf(fabsf(Pf[3] - Pf[2]), fabsf(Pf[5] - Pf[4])));
  const bool cont = dmax < F_CUTOFF;

  const float cm = cmax[cell], cn = cmin[cell];
  const float inv = 1.f / (cm - cn);
#pragma unroll
  for (int k = 0; k < 3; ++k)
    o.hl[k] = (cm * Ff[k * 2] - cn * Ff[k * 2 + 1] +
               cm * cn * (Uf[k * 2 + 1] - Uf[k * 2])) * inv;

  // classification (rho = row0, p = row1, v = row2; rho,p>0 after flip)
  const float rho0 = Pf[0], rho1 = Pf[1];
  const float p0 = Pf[2], p1 = Pf[3];
  const float v0 = Pf[4], v1 = Pf[5];
  const float c0 = sqrtf(F_GAMMA * p0 / rho0);
  const float c1 = sqrtf(F_GAMMA * p1 / rho1);
  const float dv = v1 - v0;
  const float z = 0.2f;  // (g-1)/(2g)
  const float num = c0 + c1 - 0.5f * (F_GAMMA - 1.f) * dv;
  const float den = c0 / powf(p0, z) + c1 / powf(p1, z);
  const float base = fmaxf(num / den, 1e-8f);
  const float b2 = base * base;
  const float pstar = b2 * b2 * base;  // ^(1/z) = ^5
  const bool vac = dv >= 3.f * (c0 + c1);  // 2/(g-1) = 3
  const bool dsb = pstar > fmaxf(p0, p1);
  const bool drb = pstar < fminf(p0, p1);
  o.label = vac ? 3 : (drb ? 1 : (dsb ? 0 : 2));
  o.cont = cont;
  o.flip = flip;
#pragma unroll
  for (int i = 0; i < 6; ++i) {
    o.feats[i] = Pf[i];
    o.feats[6 + i] = Uf[i];
    o.feats[12 + i] = Ff[i];
  }
  o.feats[18] = cm;
  o.feats[19] = cn;
}

// ------------------------------ main kernel --------------------------------
__global__ __launch_bounds__(128) void riemann_wmma_kernel(
    const float* __restrict__ P, const float* __restrict__ U,
    const float* __restrict__ F, const float* __restrict__ cmax,
    const float* __restrict__ cmin,
    const float* __restrict__ W1_ds, const float* __restrict__ b1_ds,
    const float* __restrict__ W2_ds, const float* __restrict__ b2_ds,
    const float* __restrict__ W1_dr, const float* __restrict__ b1_dr,
    const float* __restrict__ W2_dr, const float* __restrict__ b2_dr,
    const float* __restrict__ W1_rs, const float* __restrict__ b1_rs,
    const float* __restrict__ W2_rs, const float* __restrict__ b2_rs,
    float* __restrict__ out) {
  __shared__ __align__(16) float smem[LDS_TOT];
  const int tid = threadIdx.x;
  const int lane = tid & 31;
  const int wave = tid >> 5;

  // ---- stage weights into LDS, pre-swizzled into WMMA operand layout ----
  {
    const float* w1s[3] = {W1_ds, W1_dr, W1_rs};
    const float* b1s[3] = {b1_ds, b1_dr, b1_rs};
    const float* w2s[3] = {W2_ds, W2_dr, W2_rs};
    const float* b2s[3] = {b2_ds, b2_dr, b2_rs};
#pragma unroll
    for (int m = 0; m < 3; ++m) {
      // W1 swizzle: i = (kb*4+nb)*64 + L*2 + c  ->  W1[k][n]
#pragma unroll
      for (int it = 0; it < 10; ++it) {
        const int i = it * 128 + tid;
        const int c = i & 1;
        const int L = (i >> 1) & 31;
        const int nb = (i >> 6) & 3;
        const int kb = i >> 8;
        const int k = kb * 4 + 2 * (L >> 4) + c;
        const int n = nb * 16 + (L & 15);
        smem[L_W1SW + m * 1280 + i] = w1s[m][k * 64 + n];
      }
      // W2 swizzle: i = kb*64 + L*2 + c -> W2[k][j] (j>=3 -> 0)
#pragma unroll
      for (int it = 0; it < 8; ++it) {
        const int i = it * 128 + tid;
        const int c = i & 1;
        const int L = (i >> 1) & 31;
        const int kb = i >> 6;
        const int k = kb * 4 + 2 * (L >> 4) + c;
        const int j = L & 15;
        smem[L_W2SW + m * 1024 + i] = (j < 3) ? w2s[m][k * 3 + j] : 0.f;
      }
      if (tid < 64) smem[L_B1 + m * 64 + tid] = b1s[m][tid];
      if (tid < 3) smem[L_B2 + m * 3 + tid] = b2s[m][tid];
    }
  }
  __syncthreads();

  const int wbase = L_WAVE + wave * WV_SIZE;
  const int waveBase = (blockIdx.x * 4 + wave) * 32;
  const int cell = waveBase + lane;

  // ---- per-cell scalar phase (lane == cell-in-wave) ----
  {
    CellPre cp;
    cell_pre(P, U, F, cmax, cmin, cell, cp);
    const int fb = wbase + WV_FEAT + lane * 20;
#pragma unroll
    for (int i = 0; i < 20; ++i) smem[fb + i] = cp.feats[i];
    const int ib = wbase + WV_INFO + lane * 6;
    smem[ib + 0] = cp.hl[0];
    smem[ib + 1] = cp.hl[1];
    smem[ib + 2] = cp.hl[2];
    smem[ib + 3] = (float)cp.label;
    smem[ib + 4] = cp.cont ? 1.f : 0.f;
    smem[ib + 5] = cp.flip ? 1.f : 0.f;
  }
  asm volatile("s_wait_dscnt 0" ::: "memory");

  // WMMA f32 16x16x4 operand mapping:
  //   A 16x4: lanes 0-15 -> K0(v0)/K1(v1), lanes 16-31 -> K2(v0)/K3(v1)
  //   B 4x16: mirrored; N = lane & 15
  //   D 16x16: VGPR r -> M=r (lanes 0-15) / M=r+8 (lanes 16-31), N = lane & 15
  const int row = lane & 15;
  const int half = lane >> 4;
  const int k2 = 2 * half;

#pragma unroll
  for (int tile = 0; tile < 2; ++tile) {
    const int tbase = tile * 16;
    v8f acc2[3];

#pragma unroll
    for (int mlp = 0; mlp < 3; ++mlp) {
      // ---- layer 1: feats(16x20) @ W1(20x64), all operands ds_load_b64 ----
      v8f h0 = {}, h1 = {}, h2 = {}, h3 = {};
#pragma unroll
      for (int kb = 0; kb < 5; ++kb) {
        const int k0 = 4 * kb + k2;
        const v2f a =
            *(const v2f*)&smem[wbase + WV_FEAT + (tbase + row) * 20 + k0];
        const int wb = L_W1SW + mlp * 1280 + kb * 256 + lane * 2;
        h0 = wmma_f32(a, *(const v2f*)&smem[wb + 0 * 64], h0);
        h1 = wmma_f32(a, *(const v2f*)&smem[wb + 1 * 64], h1);
        h2 = wmma_f32(a, *(const v2f*)&smem[wb + 2 * 64], h2);
        h3 = wmma_f32(a, *(const v2f*)&smem[wb + 3 * 64], h3);
      }

      // ---- bias + tanh; store hidden pre-paired for layer-2 A operands ----
      // value h[M][col] -> htp[(col>>2)*64 + ((col>>1)&1)*32 + M*2 + (col&1)]
#pragma unroll
      for (int nb = 0; nb < 4; ++nb) {
        const v8f hh = (nb == 0) ? h0 : (nb == 1) ? h1 : (nb == 2) ? h2 : h3;
        const int col = nb * 16 + row;
        const float bv = smem[L_B1 + mlp * 64 + col];
        const int ha = wbase + WV_HTP + (col >> 2) * 64 + ((col >> 1) & 1) * 32 +
                       half * 16 + (col & 1);
#pragma unroll
        for (int r = 0; r < 8; ++r)
          smem[ha + r * 2] = fast_tanh(hh[r] + bv);
      }
      asm volatile("s_wait_dscnt 0" ::: "memory");

      // ---- layer 2: h(16x64) @ W2(64x3), all operands ds_load_b64 ----
      v8f a2 = {};
#pragma unroll
      for (int kb = 0; kb < 16; ++kb) {
        const v2f a = *(const v2f*)&smem[wbase + WV_HTP + kb * 64 + lane * 2];
        const v2f b = *(const v2f*)&smem[L_W2SW + mlp * 1024 + kb * 64 + lane * 2];
        a2 = wmma_f32(a, b, a2);
      }
      acc2[mlp] = a2;
      asm volatile("s_wait_dscnt 0" ::: "memory");
    }

    // ---- combine at data-holding lanes (N = j < 3) ----
    if (row < 3) {
      const int j = row;
      const int r0 = half * 8;
      const float b2v0 = smem[L_B2 + 0 * 3 + j];
      const float b2v1 = smem[L_B2 + 1 * 3 + j];
      const float b2v2 = smem[L_B2 + 2 * 3 + j];
#pragma unroll
      for (int r = 0; r < 8; ++r) {
        const int m = tbase + r0 + r;
        const int ib = wbase + WV_INFO + m * 6;
        const float hl = smem[ib + j];
        const int li = (int)smem[ib + 3];
        const bool cf = smem[ib + 4] > 0.5f;
        const bool ff = smem[ib + 5] > 0.5f;
        float fv = (li == 0)   ? acc2[0][r] + b2v0
                   : (li == 1) ? acc2[1][r] + b2v1
                   : (li == 2) ? acc2[2][r] + b2v2
                               : 0.f;
        if (cf) fv = hl;
        if (ff && j < 2) fv = -fv;  // flux * (-sgn) = [-1,-1,+1]
        smem[wbase + WV_FLUX + m * 3 + j] = fv;
      }
    }
    asm volatile("s_wait_dscnt 0" ::: "memory");
  }

  // ---- coalesced writeback: lane == cell-in-wave ----
  const int fb = wbase + WV_FLUX + lane * 3;
  out[cell * 3 + 0] = smem[fb + 0];
  out[cell * 3 + 1] = smem[fb + 1];
  out[cell * 3 + 2] = smem[fb + 2];
}

// ------------------------- scalar tail kernel ------------------------------
__global__ void riemann_tail_kernel(
    const float* __restrict__ P, const float* __restrict__ U,
    const float* __restrict__ F, const float* __restrict__ cmax,
    const float* __restrict__ cmin,
    const float* __restrict__ W1_ds, const float* __restrict__ b1_ds,
    const float* __restrict__ W2_ds, const float* __restrict__ b2_ds,
    const float* __restrict__ W1_dr, const float* __restrict__ b1_dr,
    const float* __restrict__ W2_dr, const float* __restrict__ b2_dr,
    const float* __restrict__ W1_rs, const float* __restrict__ b1_rs,
    const float* __restrict__ W2_rs, const float* __restrict__ b2_rs,
    float* __restrict__ out, int start, int N) {
  const int cell = start + blockIdx.x * blockDim.x + threadIdx.x;
  if (cell >= N) return;
  CellPre cp;
  cell_pre(P, U, F, cmax, cmin, cell, cp);
  const float* w1s[3] = {W1_ds, W1_dr, W1_rs};
  const float* b1s[3] = {b1_ds, b1_dr, b1_rs};
  const float* w2s[3] = {W2_ds, W2_dr, W2_rs};
  const float* b2s[3] = {b2_ds, b2_dr, b2_rs};
  float fsel[3] = {0.f, 0.f, 0.f};
  if (cp.label < 3) {
    const float* W1 = w1s[cp.label];
    const float* B1 = b1s[cp.label];
    const float* W2 = w2s[cp.label];
    const float* B2 = b2s[cp.label];
    float hbuf[64];
    for (int n = 0; n < 64; ++n) {
      float s = B1[n];
      for (int k = 0; k < 20; ++k) s += cp.feats[k] * W1[k * 64 + n];
      hbuf[n] = fast_tanh(s);
    }
    for (int j = 0; j < 3; ++j) {
      float s = B2[j];
      for (int n = 0; n < 64; ++n) s += hbuf[n] * W2[n * 3 + j];
      fsel[j] = s;
    }
  }
  for (int j = 0; j < 3; ++j) {
    float fv = cp.cont ? cp.hl[j] : fsel[j];
    if (cp.flip && j < 2) fv = -fv;
    out[cell * 3 + j] = fv;
  }
}

// ------------------------------- launcher ----------------------------------
extern "C" void kernel_launch(void* const* d_in, const int* in_sizes, int n_in,
                              void* d_out, int out_size, void* d_ws,
                              size_t ws_size, hipStream_t stream) {
  const float* P = (const float*)d_in[0];
  const float* U = (const float*)d_in[1];
  const float* F = (const float*)d_in[2];
  const float* cmax = (const float*)d_in[3];
  const float* cmin = (const float*)d_in[4];
  const float* W1_ds = (const float*)d_in[5];
  const float* b1_ds = (const float*)d_in[6];
  const float* W2_ds = (const float*)d_in[7];
  const float* b2_ds = (const float*)d_in[8];
  const float* W1_dr = (const float*)d_in[9];
  const float* b1_dr = (const float*)d_in[10];
  const float* W2_dr = (const float*)d_in[11];
  const float* b2_dr = (const float*)d_in[12];
  const float* W1_rs = (const float*)d_in[13];
  const float* b1_rs = (const float*)d_in[14];
  const float* W2_rs = (const float*)d_in[15];
  const float* b2_rs = (const float*)d_in[16];
  float* out = (float*)d_out;

  const int N = in_sizes[3];          // cmax length == number of interfaces
  const int CELLS_PER_BLOCK = 128;    // 4 waves x 32 cells
  const int nblocks = N / CELLS_PER_BLOCK;
  if (nblocks > 0) {
    riemann_wmma_kernel<<<nblocks, 128, 0, stream>>>(
        P, U, F, cmax, cmin, W1_ds, b1_ds, W2_ds, b2_ds, W1_dr, b1_dr, W2_dr,
        b2_dr, W1_rs, b1_rs, W2_rs, b2_rs, out);
  }
  const int start = nblocks * CELLS_PER_BLOCK;
  const int tail = N - start;
  if (tail > 0) {
    riemann_tail_kernel<<<(tail + 127) / 128, 128, 0, stream>>>(
        P, U, F, cmax, cmin, W1_ds, b1_ds, W2_ds, b2_ds, W1_dr, b1_dr, W2_dr,
        b2_dr, W1_rs, b1_rs, W2_rs, b2_rs, out, start, N);
  }
}